// MRR_26061861552671
// MI455X (gfx1250) — compile-verified
//
#include <hip/hip_runtime.h>
#include <hip/hip_bf16.h>
#include <stdint.h>

// MRR via rank counting. Softmax is monotone => rank(target) =
// 1 + #{v : logits[v] > logits[target]}. Pure HBM-streaming reduction:
// 823 MB @ 23.3 TB/s => ~35us floor. Bulk data is streamed with the CDNA5
// async global->LDS data mover (ASYNCcnt), non-temporal, 16B-aligned, with a
// 4-stage x 2xB128 per-lane pipeline (8 async ops in flight per wave).

#define TPB    256
#define STAGES 4
#define PER    2                      // b128 ops per stage per lane
#define CHUNK  (TPB * 4 * PER)        // floats per pipeline stage (2048)

__device__ __forceinline__ void issue_async_b128(uint32_t lds_byte_addr, uint64_t gaddr) {
  // VDST = per-lane LDS byte address, VADDR = per-lane 64-bit global address (GV mode).
  // Non-temporal: stream-once data must not thrash the 192MB L2.
  asm volatile("global_load_async_to_lds_b128 %0, %1, off th:TH_LOAD_NT"
               :: "v"(lds_byte_addr), "v"(gaddr)
               : "memory");
}

__device__ __forceinline__ void wait_async_le6() {
  asm volatile("s_wait_asynccnt 6" ::: "memory");   // 8 outstanding -> oldest stage (2 ops) done
}
__device__ __forceinline__ void wait_async_le0() {
  asm volatile("s_wait_asynccnt 0" ::: "memory");
}

__global__ __launch_bounds__(TPB)
void mrr_rank_kernel(const float* __restrict__ logits,
                     const int*   __restrict__ targets,
                     const float* __restrict__ mask,
                     float*       __restrict__ rr,
                     int V)
{
  __shared__ float4 buf[STAGES][PER][TPB];   // 32 KB, per-lane private slots
  __shared__ int    s_cnt;

  const int row = blockIdx.x;
  const int tid = threadIdx.x;
  const size_t base = (size_t)row * (size_t)V;
  const float* rowp = logits + base;

  if (tid == 0) s_cnt = 0;

  const int   t   = targets[row];
  const float tgt = rowp[t];

  int cnt = 0;

  // scalar prefix to reach 16B alignment (rows are only 4B aligned)
  const int pre = (int)(((0u - (uint32_t)(uintptr_t)rowp) & 15u) >> 2);  // 0..3
  if (tid < pre && tid < V)
    cnt += (__builtin_nontemporal_load(rowp + tid) > tgt) ? 1 : 0;

  const float* ap  = rowp + pre;              // 16B aligned
  const int    rem = V - pre;
  const int    nfull = rem / CHUNK;

  if (nfull >= STAGES) {
    const uint32_t lb = (uint32_t)(uintptr_t)(&buf[0][0][tid]);  // +8192/stage, +4096/slot
    uint64_t gcur = (uint64_t)(uintptr_t)ap + (uint64_t)tid * 16u;

    // prologue: fill all 4 stages (8 async ops in flight)
    #pragma unroll
    for (int k = 0; k < STAGES; ++k) {
      issue_async_b128(lb + (uint32_t)k * 8192u,         gcur);
      issue_async_b128(lb + (uint32_t)k * 8192u + 4096u, gcur + (uint64_t)(TPB * 16));
      gcur += (uint64_t)(CHUNK * 4);
    }

    int c = 0;
    for (; c < nfull - STAGES; ++c) {       // steady state
      wait_async_le6();
      const int s = c & (STAGES - 1);
      float4 a = buf[s][0][tid];
      float4 b = buf[s][1][tid];
      cnt += (a.x > tgt) + (a.y > tgt) + (a.z > tgt) + (a.w > tgt);
      cnt += (b.x > tgt) + (b.y > tgt) + (b.z > tgt) + (b.w > tgt);
      issue_async_b128(lb + (uint32_t)s * 8192u,         gcur);
      issue_async_b128(lb + (uint32_t)s * 8192u + 4096u, gcur + (uint64_t)(TPB * 16));
      gcur += (uint64_t)(CHUNK * 4);
    }
    wait_async_le0();                        // drain
    for (; c < nfull; ++c) {
      const int s = c & (STAGES - 1);
      float4 a = buf[s][0][tid];
      float4 b = buf[s][1][tid];
      cnt += (a.x > tgt) + (a.y > tgt) + (a.z > tgt) + (a.w > tgt);
      cnt += (b.x > tgt) + (b.y > tgt) + (b.z > tgt) + (b.w > tgt);
    }
    // tail (rem % CHUNK elements), coalesced scalar NT loads
    for (int i = nfull * CHUNK + tid; i < rem; i += TPB)
      cnt += (__builtin_nontemporal_load(ap + i) > tgt) ? 1 : 0;
  } else {
    for (int i = tid; i < rem; i += TPB)
      cnt += (__builtin_nontemporal_load(ap + i) > tgt) ? 1 : 0;
  }

  __syncthreads();
  atomicAdd(&s_cnt, cnt);
  __syncthreads();
  if (tid == 0)
    rr[row] = mask[row] / (float)(s_cnt + 1);   // mask * 1/rank
}

__global__ __launch_bounds__(TPB)
void mrr_finalize_kernel(const float* __restrict__ rr,
                         const float* __restrict__ mask,
                         float* __restrict__ out,
                         int R)
{
  __shared__ float s_num[TPB];
  __shared__ float s_den[TPB];
  const int tid = threadIdx.x;

  float a = 0.f, b = 0.f;
  for (int i = tid; i < R; i += TPB) { a += rr[i]; b += mask[i]; }
  s_num[tid] = a; s_den[tid] = b;
  __syncthreads();
  for (int s = TPB / 2; s > 0; s >>= 1) {
    if (tid < s) { s_num[tid] += s_num[tid + s]; s_den[tid] += s_den[tid + s]; }
    __syncthreads();
  }
  if (tid == 0) out[0] = s_num[0] / s_den[0];
}

extern "C" void kernel_launch(void* const* d_in, const int* in_sizes, int n_in,
                              void* d_out, int out_size, void* d_ws, size_t ws_size,
                              hipStream_t stream) {
  const float* logits  = (const float*)d_in[0];
  const int*   targets = (const int*)  d_in[1];
  const float* mask    = (const float*)d_in[2];
  float*       out     = (float*)d_out;

  const int R = in_sizes[1];          // B*S rows (4096)
  const int V = in_sizes[0] / R;      // vocab (50257)
  float* rr = (float*)d_ws;           // R floats of scratch, fully overwritten

  mrr_rank_kernel<<<R, TPB, 0, stream>>>(logits, targets, mask, rr, V);
  mrr_finalize_kernel<<<1, TPB, 0, stream>>>(rr, mask, out, R);
}